// simple_gcn_84026740179211
// MI455X (gfx1250) — compile-verified
//
#include <hip/hip_runtime.h>
#include <hip/hip_bf16.h>
#include <hip/amd_detail/amd_hip_unsafe_atomics.h>

typedef float v2f __attribute__((ext_vector_type(2)));
typedef float v8f __attribute__((ext_vector_type(8)));

#define D 64

// ---------------- degree / norm kernels ----------------

__global__ void k_init_deg(float* deg, int n) {
    int i = blockIdx.x * blockDim.x + threadIdx.x;
    if (i < n) deg[i] = 1.0f;  // self loop
}

__global__ void k_count_deg(const int* col, float* deg, int e) {
    int i = blockIdx.x * blockDim.x + threadIdx.x;
    if (i < e) unsafeAtomicAdd(&deg[col[i]], 1.0f);
}

__global__ void k_rsqrt_deg(float* deg, int n) {
    int i = blockIdx.x * blockDim.x + threadIdx.x;
    if (i < n) deg[i] = rsqrtf(deg[i]);  // in-place: deg -> dinv
}

// ---------------- WMMA GEMM: h = xcat @ W^T, fused epilogue ----------------
// Each wave computes a 16x64 tile of h. A = xcat (16xK), B = W^T staged in LDS
// interleaved by K-pair so each B fragment is one aligned ds_load_b64 into an
// adjacent (even-aligned) VGPR pair -- no shuffle moves before v_wmma.
// Epilogue: h -> workspace; out = bias + h * dinv^2 (self-loop term).

__global__ __launch_bounds__(256) void k_gemm_wmma(
    const float* __restrict__ x,      // [N,61]
    const float* __restrict__ t3,     // [N,3]
    const float* __restrict__ W,      // [64,64] layer-4 weight, row-major [out,in]
    const float* __restrict__ bias,   // [64]
    const float* __restrict__ dinv,   // [N]
    float* __restrict__ hbuf,         // [N,64]
    float* __restrict__ out,          // [N,64]
    int n)
{
    // ldsP[p*128 + nn*2 + s] = B[2p+s][nn] = W[nn][2p+s]   (p = k/2, s in {0,1})
    __shared__ __align__(16) float ldsP[D * D];
    __shared__ float ldsBias[D];

    int tid = threadIdx.x;
    // stage W^T (K-pair interleaved) and bias into LDS; float2 granularity
    for (int i2 = tid; i2 < (D * D) / 2; i2 += 256) {
        int nn = i2 & 63;           // output column
        int p  = i2 >> 6;           // k-pair
        v2f w = *(const v2f*)&W[nn * D + 2 * p];   // {W[nn][2p], W[nn][2p+1]}
        *(v2f*)&ldsP[p * 128 + nn * 2] = w;
    }
    if (tid < D) ldsBias[tid] = bias[tid];
    __syncthreads();

    int wave = tid >> 5;
    int lane = tid & 31;
    int tile = blockIdx.x * 8 + wave;       // 16-row tile index (wave-uniform)
    int r0 = tile * 16;
    if (r0 >= n) return;                    // wave-uniform: EXEC stays full

    int m  = lane & 15;                     // A row / B col / C col within tile
    int hh = lane >> 4;                     // half-wave select
    int kb = hh << 1;                       // K offset {0,2} within 4-wide K step
    int arow = r0 + m;

    v8f acc0 = {}, acc1 = {}, acc2 = {}, acc3 = {};

#pragma unroll
    for (int kk = 0; kk < D; kk += 4) {
        int c0 = kk + kb;
        int c1 = c0 + 1;
        v2f a;
        a[0] = (c0 < 61) ? x[arow * 61 + c0] : t3[arow * 3 + (c0 - 61)];
        a[1] = (c1 < 61) ? x[arow * 61 + c1] : t3[arow * 3 + (c1 - 61)];

        // pair index for this lane's K slice
        const float* Bp = &ldsP[((kk >> 1) + hh) * 128];
        v2f b0 = *(const v2f*)&Bp[(m)      * 2];
        v2f b1 = *(const v2f*)&Bp[(16 + m) * 2];
        v2f b2 = *(const v2f*)&Bp[(32 + m) * 2];
        v2f b3 = *(const v2f*)&Bp[(48 + m) * 2];

        acc0 = __builtin_amdgcn_wmma_f32_16x16x4_f32(false, a, false, b0, (short)0, acc0, false, false);
        acc1 = __builtin_amdgcn_wmma_f32_16x16x4_f32(false, a, false, b1, (short)0, acc1, false, false);
        acc2 = __builtin_amdgcn_wmma_f32_16x16x4_f32(false, a, false, b2, (short)0, acc2, false, false);
        acc3 = __builtin_amdgcn_wmma_f32_16x16x4_f32(false, a, false, b3, (short)0, acc3, false, false);
    }

    // Epilogue per C/D layout: VGPR j -> row r0 + hh*8 + j, col = (lane&15) + 16*ntile
#pragma unroll
    for (int j = 0; j < 8; ++j) {
        int rr = r0 + hh * 8 + j;
        float dv = dinv[rr];
        float sn = dv * dv;               // self-loop norm = 1/deg
        int base = rr * D;
        float h0 = acc0[j], h1 = acc1[j], h2 = acc2[j], h3 = acc3[j];
        hbuf[base + m]      = h0;  out[base + m]      = ldsBias[m]      + h0 * sn;
        hbuf[base + 16 + m] = h1;  out[base + 16 + m] = ldsBias[16 + m] + h1 * sn;
        hbuf[base + 32 + m] = h2;  out[base + 32 + m] = ldsBias[32 + m] + h2 * sn;
        hbuf[base + 48 + m] = h3;  out[base + 48 + m] = ldsBias[48 + m] + h3 * sn;
    }
}

// ---------------- edge scatter: out[col] += h[row] * dinv[row]*dinv[col] ----------------

__global__ __launch_bounds__(256) void k_scatter(
    const int* __restrict__ row, const int* __restrict__ col,
    const float* __restrict__ hbuf, const float* __restrict__ dinv,
    float* __restrict__ out, int e)
{
    int tid = threadIdx.x;
    int eid = blockIdx.x * 4 + (tid >> 6);   // 4 edges per 256-thread block
    if (eid >= e) return;
    int c = tid & 63;
    int r = row[eid];
    int t = col[eid];
    float nrm = dinv[r] * dinv[t];
    float val = hbuf[r * D + c] * nrm;
    unsafeAtomicAdd(&out[t * D + c], val);
}

// ---------------- relu ----------------

__global__ void k_relu(float* out, int total) {
    int i = blockIdx.x * blockDim.x + threadIdx.x;
    if (i < total) out[i] = fmaxf(out[i], 0.0f);
}

// ---------------- launch ----------------

extern "C" void kernel_launch(void* const* d_in, const int* in_sizes, int n_in,
                              void* d_out, int out_size, void* d_ws, size_t ws_size,
                              hipStream_t stream) {
    const float* x   = (const float*)d_in[0];   // [N,61]
    const float* t3  = (const float*)d_in[1];   // [N,3]
    const int*   ei  = (const int*)d_in[2];     // [2,E]
    const float* Ws  = (const float*)d_in[3];   // [5,64,64]
    const float* bs  = (const float*)d_in[4];   // [5,64]

    int n = in_sizes[0] / 61;
    int e = in_sizes[2] / 2;
    int depth = in_sizes[3] / (D * D);
    const float* W4 = Ws + (size_t)(depth - 1) * D * D;
    const float* b4 = bs + (size_t)(depth - 1) * D;
    const int* row = ei;
    const int* col = ei + e;

    float* dinv = (float*)d_ws;                 // [N]
    float* hbuf = dinv + n;                     // [N,64]
    float* out  = (float*)d_out;                // [N,64]

    k_init_deg<<<(n + 255) / 256, 256, 0, stream>>>(dinv, n);
    k_count_deg<<<(e + 255) / 256, 256, 0, stream>>>(col, dinv, e);
    k_rsqrt_deg<<<(n + 255) / 256, 256, 0, stream>>>(dinv, n);

    int tiles = (n + 15) / 16;                  // N=100000 -> 6250 full tiles
    int gemm_blocks = (tiles + 7) / 8;
    k_gemm_wmma<<<gemm_blocks, 256, 0, stream>>>(x, t3, W4, b4, dinv, hbuf, out, n);

    k_scatter<<<(e + 3) / 4, 256, 0, stream>>>(row, col, hbuf, dinv, out, e);

    k_relu<<<(n * D + 255) / 256, 256, 0, stream>>>(out, (int)((size_t)n * D));
}